// SlidingWindowInferenceWrapper_1554778161720
// MI455X (gfx1250) — compile-verified
//
#include <hip/hip_runtime.h>
#include <stdint.h>

// Batched NMS for MI455X (gfx1250, wave32).
// Kernel A: per-batch radix top-k (K=1000 of N=400000) + LDS bitonic sort.
// Kernel B: per-batch NMS: LDS-resident boxes, WMMA f32 16x16x4 computes the
//           pairwise area-sum (union denominator) per 16x16 tile, VALU does
//           intersection; suppression decided divide-free (inter > THR*union);
//           bits packed via wave32 ballot into a 1024x1024 LDS bit-matrix;
//           one wave runs the serial greedy loop as 32-lane bitwise ANDs.

#define BATCH   16
#define TOPK    1000
#define KPAD    1024
#define MAXPRED 300
#define IOU_THR 0.65f
#define CAND_CAP 2048

typedef __attribute__((ext_vector_type(2))) float v2f;
typedef __attribute__((ext_vector_type(8))) float v8f;

// ---------------------------------------------------------------------------
// Kernel A: top-k select. One 1024-thread workgroup per batch.
// Scores are uniform [0,1) floats -> raw bits order like values.
// 3 radix passes (8 bits each) find a 24-bit threshold, then candidates are
// compacted to LDS and bitonic-sorted descending by (score_bits, ~index).
// ---------------------------------------------------------------------------
__global__ __launch_bounds__(1024, 1)
void topk_kernel(const float* __restrict__ scores, int N,
                 float* __restrict__ ws_score, int* __restrict__ ws_idx)
{
    __shared__ unsigned int hist[256];
    __shared__ unsigned long long cand[CAND_CAP];
    __shared__ unsigned int s_prefix, s_remaining, s_count;

    const int b   = blockIdx.x;
    const int tid = threadIdx.x;
    const float* sc = scores + (size_t)b * N;

    if (tid == 0) { s_prefix = 0u; s_remaining = TOPK; }
    __syncthreads();

    for (int pass = 0; pass < 3; ++pass) {
        for (int i = tid; i < 256; i += 1024) hist[i] = 0u;
        __syncthreads();
        const unsigned int prefix = s_prefix;
        const int shift = 24 - 8 * pass;
        for (int i = tid; i < N; i += 1024) {
            __builtin_prefetch(sc + i + 8192, 0, 0);
            unsigned int key = __float_as_uint(sc[i]);
            bool in = (pass == 0) || ((key >> (shift + 8)) == prefix);
            if (in) atomicAdd(&hist[(key >> shift) & 0xFFu], 1u);
        }
        __syncthreads();
        if (tid == 0) {
            unsigned int rem = s_remaining, cum = 0u, bucket = 0u;
            for (int bkt = 255; bkt >= 0; --bkt) {
                if (cum + hist[bkt] >= rem) { bucket = (unsigned)bkt; s_remaining = rem - cum; break; }
                cum += hist[bkt];
            }
            s_prefix = (prefix << 8) | bucket;
        }
        __syncthreads();
    }

    const unsigned int thr24 = s_prefix;
    if (tid == 0) s_count = 0u;
    __syncthreads();

    // collect all elements with top-24 bits >= threshold (guaranteed >= TOPK)
    for (int i = tid; i < N; i += 1024) {
        unsigned int key = __float_as_uint(sc[i]);
        if ((key >> 8) >= thr24) {
            unsigned int slot = atomicAdd(&s_count, 1u);
            if (slot < CAND_CAP)
                cand[slot] = ((unsigned long long)key << 32) | (unsigned int)(~(unsigned)i);
        }
    }
    __syncthreads();
    unsigned int cnt = s_count; if (cnt > CAND_CAP) cnt = CAND_CAP;
    for (int i = (int)cnt + tid; i < CAND_CAP; i += 1024) cand[i] = 0ull;
    __syncthreads();

    // bitonic sort, descending (ties: smaller original index first via ~idx)
    for (int k = 2; k <= CAND_CAP; k <<= 1) {
        for (int j = k >> 1; j > 0; j >>= 1) {
            for (int i = tid; i < CAND_CAP; i += 1024) {
                int l = i ^ j;
                if (l > i) {
                    unsigned long long a = cand[i], c = cand[l];
                    bool descRegion = ((i & k) == 0);
                    if (descRegion ? (a < c) : (a > c)) { cand[i] = c; cand[l] = a; }
                }
            }
            __syncthreads();
        }
    }

    unsigned long long v = cand[tid];
    ws_score[b * KPAD + tid] = __uint_as_float((unsigned int)(v >> 32));
    ws_idx  [b * KPAD + tid] = (int)(~(unsigned int)(v & 0xFFFFFFFFull));
}

// ---------------------------------------------------------------------------
// Kernel B: NMS. One 1024-thread (32-wave) workgroup per batch.
// LDS: class-offset boxes SoA (+areas), original dets, 1024x1024 suppression
// bit-matrix (128 KB), keep mask + prefix. ~180 KB total (< 320 KB/WGP).
// ---------------------------------------------------------------------------
__global__ __launch_bounds__(1024, 1)
void nms_kernel(const float* __restrict__ boxes, const int* __restrict__ classes,
                const float* __restrict__ ws_score, const int* __restrict__ ws_idx,
                float* __restrict__ out, int N)
{
    extern __shared__ char smem[];
    float* sx1   = (float*)smem;          // KPAD
    float* sy1   = sx1 + KPAD;
    float* sx2   = sy1 + KPAD;
    float* sy2   = sx2 + KPAD;
    float* sarea = sy2 + KPAD;
    float* sox   = sarea + KPAD;          // 4*KPAD original boxes (AoS)
    float* sscore= sox + 4 * KPAD;
    float* scls  = sscore + KPAD;
    float* sred  = scls + KPAD;           // KPAD reduction scratch
    unsigned int* ssup  = (unsigned int*)(sred + KPAD);   // KPAD * 32 words
    unsigned int* skeep = ssup + KPAD * 32;               // 32
    unsigned int* spref = skeep + 32;                     // 33

    const int b   = blockIdx.x;
    const int tid = threadIdx.x;

    // ---- gather top-k dets, find max coordinate -----------------------------
    float lx1 = 0.f, ly1 = 0.f, lx2 = 0.f, ly2 = 0.f, sc = 0.f;
    int   cl  = 0;
    float lmax = -1e30f;
    if (tid < TOPK) {
        int idx = ws_idx[b * KPAD + tid];
        sc = ws_score[b * KPAD + tid];
        const float* bp = boxes + ((size_t)b * N + idx) * 4;
        lx1 = bp[0]; ly1 = bp[1]; lx2 = bp[2]; ly2 = bp[3];
        cl  = classes[(size_t)b * N + idx];
        lmax = fmaxf(fmaxf(lx1, ly1), fmaxf(lx2, ly2));
    }
    sred[tid] = lmax;
    __syncthreads();
    for (int s = 512; s > 0; s >>= 1) {
        if (tid < s) sred[tid] = fmaxf(sred[tid], sred[tid + s]);
        __syncthreads();
    }
    const float maxc = sred[0] + 1.0f;

    // ---- class-offset boxes + areas into LDS (SoA) --------------------------
    const float off = (float)cl * maxc;
    const float bx1 = lx1 + off, by1 = ly1 + off, bx2 = lx2 + off, by2 = ly2 + off;
    sx1[tid] = bx1; sy1[tid] = by1; sx2[tid] = bx2; sy2[tid] = by2;
    sarea[tid] = (bx2 - bx1) * (by2 - by1);
    sox[tid * 4 + 0] = lx1; sox[tid * 4 + 1] = ly1;
    sox[tid * 4 + 2] = lx2; sox[tid * 4 + 3] = ly2;
    sscore[tid] = sc; scls[tid] = (float)cl;

    for (int i = tid; i < KPAD * 32; i += 1024) ssup[i] = 0u;
    __syncthreads();

    // ---- suppression bit-matrix via WMMA tiles ------------------------------
    const int lane = tid & 31;
    const int wv   = tid >> 5;       // wave 0..31
    const int half = lane >> 4;      // 0 or 1
    const int ln   = lane & 15;
    const float hsel = (half == 0) ? 1.0f : 0.0f;   // zero out K=2,3 lanes

    for (int t = wv; t < 64 * 64; t += 32) {
        const int ti = t >> 6, tj = t & 63;
        if (tj < ti) continue;                     // upper-triangle tiles incl. diagonal only
        const int i0 = ti * 16, j0 = tj * 16;

        // rank-2 matmul: D[M][N] = area_i[M]*1 + 1*area_j[N]
        // (unconditional LDS loads; select by half afterwards - no EXEC churn)
        const float ai = sarea[i0 + ln];
        const float aj = sarea[j0 + ln];
        v2f amat, bmat;
        amat.x = ai * hsel;   // A[M][K=0]
        amat.y = hsel;        // A[M][K=1]
        bmat.x = hsel;        // B[K=0][N]
        bmat.y = aj * hsel;   // B[K=1][N]
        v8f csum = {};
        csum = __builtin_amdgcn_wmma_f32_16x16x4_f32(
            false, amat, false, bmat, (short)0, csum, false, false);

        const int j = j0 + ln;                    // this lane's column
        const float jx1 = sx1[j], jy1 = sy1[j], jx2 = sx2[j], jy2 = sy2[j];

        #pragma unroll
        for (int m = 0; m < 8; ++m) {
            const int i = i0 + m + half * 8;      // row per C/D VGPR layout
            float iw = fminf(sx2[i], jx2) - fmaxf(sx1[i], jx1);
            float ih = fminf(sy2[i], jy2) - fmaxf(sy1[i], jy1);
            float inter = fmaxf(iw, 0.0f) * fmaxf(ih, 0.0f);
            float uni   = csum[m] - inter + 1e-9f;
            // union > 0 always => iou > THR  <=>  inter > THR * union
            bool  supv  = inter > IOU_THR * uni;
            unsigned int bal = __builtin_amdgcn_ballot_w32(supv);
            // lanes m / 16+m commit rows i0+m / i0+m+8 (cols j0..j0+15)
            if (lane == m || lane == 16 + m) {
                const int row = i0 + m + half * 8;
                unsigned int bits = half ? ((bal >> 16) & 0xFFFFu) : (bal & 0xFFFFu);
                const int d = row - j0;           // keep only columns j > row
                unsigned int mask = 0xFFFFu;
                if (d >= 15)     mask = 0u;
                else if (d >= 0) mask = 0xFFFFu & ~((2u << d) - 1u);
                bits &= mask;
                if (bits) atomicOr(&ssup[row * 32 + (j0 >> 5)], bits << (j0 & 31));
            }
        }
    }
    __syncthreads();

    // ---- serial greedy suppression: one wave, bitwise -----------------------
    if (wv == 0) {
        unsigned int keepw = 0xFFFFFFFFu;         // lane owns word `lane`
        for (int i = 0; i < TOPK; ++i) {
            unsigned int kw = (unsigned int)__shfl((int)keepw, i >> 5, 32);
            if ((kw >> (i & 31)) & 1u)
                keepw &= ~ssup[i * 32 + lane];
        }
        if (lane == 31) keepw &= 0xFFu;           // clear pad bits 1000..1023
        skeep[lane] = keepw;
    }
    __syncthreads();

    if (tid < 32) {
        unsigned int p = 0u;
        for (int w = 0; w < tid; ++w) p += (unsigned)__popc(skeep[w]);
        spref[tid] = p;
        if (tid == 31) spref[32] = p + (unsigned)__popc(skeep[31]);
    }
    __syncthreads();

    // ---- emit: kept dets in score order, zero-fill the rest -----------------
    const unsigned int totalKept = spref[32];
    if (tid < TOPK) {
        const unsigned int w = (unsigned)tid >> 5, bit = (unsigned)tid & 31u;
        if ((skeep[w] >> bit) & 1u) {
            unsigned int pos = spref[w] + (unsigned)__popc(skeep[w] & ((1u << bit) - 1u));
            if (pos < MAXPRED) {
                float* o = out + ((size_t)b * MAXPRED + pos) * 6;
                o[0] = sox[tid * 4 + 0]; o[1] = sox[tid * 4 + 1];
                o[2] = sox[tid * 4 + 2]; o[3] = sox[tid * 4 + 3];
                o[4] = sscore[tid];      o[5] = scls[tid];
            }
        }
    }
    if (tid < MAXPRED && (unsigned)tid >= totalKept) {
        float* o = out + ((size_t)b * MAXPRED + tid) * 6;
        o[0] = o[1] = o[2] = o[3] = o[4] = o[5] = 0.0f;
    }
}

// ---------------------------------------------------------------------------
extern "C" void kernel_launch(void* const* d_in, const int* in_sizes, int n_in,
                              void* d_out, int out_size, void* d_ws, size_t ws_size,
                              hipStream_t stream)
{
    const float* boxes   = (const float*)d_in[0];
    const float* scores  = (const float*)d_in[1];
    const int*   classes = (const int*)d_in[2];
    float*       out     = (float*)d_out;
    const int N = in_sizes[1] / BATCH;

    float* ws_score = (float*)d_ws;
    int*   ws_idx   = (int*)((char*)d_ws + (size_t)BATCH * KPAD * sizeof(float));

    topk_kernel<<<BATCH, 1024, 0, stream>>>(scores, N, ws_score, ws_idx);

    const size_t smemB =
        (size_t)(12 * KPAD) * sizeof(float) +            // boxes SoA + dets + scratch
        (size_t)KPAD * 32 * sizeof(unsigned int) +       // suppression bit-matrix
        (size_t)(32 + 33) * sizeof(unsigned int) + 16;   // keep mask + prefix
    nms_kernel<<<BATCH, 1024, smemB, stream>>>(boxes, classes, ws_score, ws_idx, out, N);
}